// PGNN_layer_51694226375361
// MI455X (gfx1250) — compile-verified
//
#include <hip/hip_runtime.h>

#define NPTS 20000
#define KNN  32
#define DIN  128
#define DOUT 128

typedef float v2f __attribute__((ext_vector_type(2)));
typedef float v8f __attribute__((ext_vector_type(8)));

// ---------------------------------------------------------------------------
// Kernel 1: P = data_x @ wh[:, :128]^T, Q = data_x @ wh[:, 128:]^T
// One wave per 16x16 output tile, fp32 WMMA (V_WMMA_F32_16X16X4_F32),
// K=128 accumulated as 32 chained WMMAs.
// wh is row-major [128, 256]: W[i, o'] = wh[o'*256 + i]        (o' <  128 -> P)
//                             W[i, o'] = wh[(o'-128)*256+128+i] (o' >= 128 -> Q)
// ---------------------------------------------------------------------------
__global__ __launch_bounds__(512)
void pgnn_precompute_gemm(const float* __restrict__ data_x,
                          const float* __restrict__ wh,
                          float* __restrict__ P,
                          float* __restrict__ Q)
{
    const int lane    = threadIdx.x & 31;
    const int ct      = threadIdx.x >> 5;   // column tile 0..15 (one per wave)
    const int tile    = blockIdx.x;         // row tile 0..1249 (20000/16)
    const int halfsel = lane >> 4;          // 0: K pair {0,1}, 1: K pair {2,3}
    const int l15     = lane & 15;

    // A fragment source: row = tile*16 + l15, consecutive K pair per lane
    const float* aptr = data_x + (size_t)(tile * 16 + l15) * DIN + halfsel * 2;

    // B fragment source: column o' of W, consecutive K pair (contiguous in wh)
    const int ocol = ct * 16 + l15;         // 0..255
    const float* bptr = (ocol < DOUT)
        ? (wh + (size_t)ocol * (2 * DIN) + halfsel * 2)
        : (wh + (size_t)(ocol - DOUT) * (2 * DIN) + DIN + halfsel * 2);

    v8f c = {0.f, 0.f, 0.f, 0.f, 0.f, 0.f, 0.f, 0.f};
    #pragma unroll
    for (int k = 0; k < DIN; k += 4) {
        v2f a = *(const v2f*)(aptr + k);
        v2f b = *(const v2f*)(bptr + k);
        // 8 args: (neg_a, A, neg_b, B, c_mod, C, reuse_a, reuse_b)
        c = __builtin_amdgcn_wmma_f32_16x16x4_f32(false, a, false, b,
                                                  (short)0, c, false, false);
    }

    // C/D layout: VGPR v -> row (v + 8*halfsel), col = l15 within tile
    const int mbase = tile * 16 + halfsel * 8;
    float* outp = (ocol < DOUT) ? (P + ocol) : (Q + (ocol - DOUT));
    #pragma unroll
    for (int v = 0; v < 8; ++v)
        outp[(size_t)(mbase + v) * DOUT] = c[v];
}

// ---------------------------------------------------------------------------
// Kernel 2: fused d-MLP + gather + relu + epilogue. One wave32 per node n.
// Lane owns channels [lane*4, lane*4+4) -> b128 loads/stores, L2-resident
// gathers of P rows. h never hits memory.
// ---------------------------------------------------------------------------
__device__ __forceinline__ float wave_sum(float v) {
    #pragma unroll
    for (int off = 16; off > 0; off >>= 1)
        v += __shfl_xor(v, off, 32);
    return v;
}

__global__ __launch_bounds__(256)
void pgnn_fused(const float* __restrict__ dists_max,
                const long long* __restrict__ dists_argmax,
                const float* __restrict__ w1d, const float* __restrict__ b1d,
                const float* __restrict__ w2d, const float* __restrict__ b2d,
                const float* __restrict__ bh,  const float* __restrict__ wp,
                const float* __restrict__ bp,
                const float* __restrict__ P,   const float* __restrict__ Q,
                float* __restrict__ out_pos,   float* __restrict__ out_struct)
{
    const int lane = threadIdx.x & 31;
    const int wv   = threadIdx.x >> 5;
    const int n    = blockIdx.x * 8 + wv;
    if (n >= NPTS) return;

    const int c0 = lane * 4;
    const float4 q   = *(const float4*)(Q   + (size_t)n * DOUT + c0);
    const float4 bhv = *(const float4*)(bh  + c0);
    const float4 w1  = *(const float4*)(w1d + c0);
    const float4 b1  = *(const float4*)(b1d + c0);
    const float4 w2  = *(const float4*)(w2d + c0);
    const float4 wpv = *(const float4*)(wp  + c0);
    const float  b2  = b2d[0];
    const float  bps = bp[0];

    // fold Q + bh once
    const float4 qb = make_float4(q.x + bhv.x, q.y + bhv.y,
                                  q.z + bhv.z, q.w + bhv.w);

    float4 acc = make_float4(0.f, 0.f, 0.f, 0.f);

    for (int k = 0; k < KNN; ++k) {
        const float     x   = dists_max[(size_t)n * KNN + k];
        const long long idx = dists_argmax[(size_t)n * KNN + k];

        // d[n,k] = sum_o relu(x*w1d[o] + b1d[o]) * w2d[o] + b2d
        float t = fmaxf(fmaf(x, w1.x, b1.x), 0.f) * w2.x
                + fmaxf(fmaf(x, w1.y, b1.y), 0.f) * w2.y
                + fmaxf(fmaf(x, w1.z, b1.z), 0.f) * w2.z
                + fmaxf(fmaf(x, w1.w, b1.w), 0.f) * w2.w;
        const float dd = wave_sum(t) + b2;

        // h = relu(d * P[idx] + Q[n] + bh)   (L2-resident gather: P is 10 MB)
        const float4 pg = *(const float4*)(P + (size_t)idx * DOUT + c0);
        float4 h;
        h.x = fmaxf(fmaf(dd, pg.x, qb.x), 0.f);
        h.y = fmaxf(fmaf(dd, pg.y, qb.y), 0.f);
        h.z = fmaxf(fmaf(dd, pg.z, qb.z), 0.f);
        h.w = fmaxf(fmaf(dd, pg.w, qb.w), 0.f);

        acc.x += h.x; acc.y += h.y; acc.z += h.z; acc.w += h.w;

        float p = h.x * wpv.x + h.y * wpv.y + h.z * wpv.z + h.w * wpv.w;
        p = wave_sum(p);
        if (lane == 0)
            out_pos[(size_t)n * KNN + k] = p + bps;
    }

    const float inv_k = 1.0f / (float)KNN;
    float4 os = make_float4(acc.x * inv_k, acc.y * inv_k,
                            acc.z * inv_k, acc.w * inv_k);
    *(float4*)(out_struct + (size_t)n * DOUT + c0) = os;
}

// ---------------------------------------------------------------------------
extern "C" void kernel_launch(void* const* d_in, const int* in_sizes, int n_in,
                              void* d_out, int out_size, void* d_ws, size_t ws_size,
                              hipStream_t stream)
{
    const float*     data_x       = (const float*)d_in[0];
    const float*     dists_max    = (const float*)d_in[1];
    const long long* dists_argmax = (const long long*)d_in[2];
    const float*     w1d          = (const float*)d_in[3];
    const float*     b1d          = (const float*)d_in[4];
    const float*     w2d          = (const float*)d_in[5];
    const float*     b2d          = (const float*)d_in[6];
    const float*     wh           = (const float*)d_in[7];
    const float*     bh           = (const float*)d_in[8];
    const float*     wp           = (const float*)d_in[9];
    const float*     bp           = (const float*)d_in[10];

    float* P = (float*)d_ws;                    // [N, 128]  10.24 MB
    float* Q = P + (size_t)NPTS * DOUT;         // [N, 128]  10.24 MB

    float* out_pos    = (float*)d_out;                       // [N, K]
    float* out_struct = out_pos + (size_t)NPTS * KNN;        // [N, 128]

    pgnn_precompute_gemm<<<NPTS / 16, 512, 0, stream>>>(data_x, wh, P, Q);
    pgnn_fused<<<(NPTS + 7) / 8, 256, 0, stream>>>(
        dists_max, dists_argmax, w1d, b1d, w2d, b2d, bh, wp, bp,
        P, Q, out_pos, out_struct);
}